// CTCLoss_6571299963200
// MI455X (gfx1250) — compile-verified
//
#include <hip/hip_runtime.h>
#include <hip/hip_bf16.h>

// CTC forward loss for MI455X (gfx1250, wave32).
// One 256-thread block (8 waves) per batch element, one thread per extended
// state (L=201). Alpha ping-pongs in LDS with 2 NEG guard slots per buffer so
// neighbor reads are branch-free (merge to ds_load_2addr). One split barrier
// per DP step (s_barrier_signal/-wait -1) with the wave-private
// s_wait_asynccnt overlapped between signal and wait. Emission rows stream
// through a 4-deep LDS ring via CDNA5 async global->LDS B128 copies.
// The 3-way recurrence is a branch-free logsumexp (v_max3 + 3x v_exp_f32 +
// 1x v_log_f32) with the skip path masked to NEG (exact).
// The scan terminates at te = input_length-1: beyond that alpha is frozen in
// the reference, so those steps are no-ops (il is uniform per block).

#define WAIT_ASYNCCNT(n) asm volatile("s_wait_asynccnt %0" :: "n"(n) : "memory")
#define WAIT_DSCNT(n)    asm volatile("s_wait_dscnt %0"    :: "n"(n) : "memory")
#define BAR_SIGNAL()     asm volatile("s_barrier_signal -1" ::: "memory")
#define BAR_WAIT()       asm volatile("s_barrier_wait -1"   ::: "memory")

constexpr int   T    = 1000;
constexpr int   B    = 32;
constexpr int   C    = 1000;
constexpr int   S    = 100;
constexpr int   NT   = 256;    // threads per block; state s = tid
constexpr int   PF   = 4;      // LDS ring depth (rows in flight per wave)
constexpr int   ROWF = 1024;   // floats per LDS row slot (>= C)
constexpr int   AP   = NT + 2; // alpha buffer pitch (2 NEG guard slots)
constexpr float NEGV = -1e30f;

__device__ __forceinline__ float lae(float a, float b) {
    float m = fmaxf(a, b);
    float d = fminf(a, b) - m;
    return m + __logf(1.0f + __expf(d));
}

// Branch-free 3-way logsumexp; NEG inputs contribute exp(-huge)=0 exactly,
// all-NEG gives NEGV + log(3) == NEGV in fp32 (matches jnp semantics).
__device__ __forceinline__ float lae3(float a, float b, float c) {
    float m = fmaxf(fmaxf(a, b), c);                  // v_max3_f32
    float e = __expf(a - m) + __expf(b - m) + __expf(c - m);
    return m + __logf(e);
}

// Each thread async-copies its 16 B of one emission row (4000 B): one
// global_load_async_to_lds_b128 per wave per row. Offsets clamped to 3984 so
// clamped lanes duplicate the final 16 B (same data, never OOB).
__device__ __forceinline__ void async_row_tile(const float* __restrict__ row,
                                               const float* ldsrow, unsigned off) {
    unsigned lds = (unsigned)(unsigned long long)ldsrow + off;
    asm volatile("global_load_async_to_lds_b128 %0, %1, %2"
                 :: "v"(lds), "v"(off), "s"(row)
                 : "memory");
}

__global__ __launch_bounds__(NT)
void ctc_alpha_kernel(const float* __restrict__ log_probs,
                      const int*   __restrict__ targets,
                      const int*   __restrict__ input_lengths,
                      const int*   __restrict__ target_lengths,
                      float*       __restrict__ partial) {
    __shared__ float rows[PF * ROWF];     // emission-row ring (16 KB)
    __shared__ float shA[2 * AP];         // alpha ping-pong + guards

    const int b   = blockIdx.x;
    const int tid = threadIdx.x;
    const int s   = tid;
    const int tl  = target_lengths[b];
    const int il  = input_lengths[b];

    // last step that actually updates alpha (uniform per block)
    const int te    = (il - 1) < (T - 1) ? (il - 1) : (T - 1);
    const int tmain = te < (T - PF) ? te : (T - PF);

    // loop-invariant async copy offset for this thread
    unsigned off = (unsigned)(tid * 16);
    if (off > 3984u) off = 3984u;

    // ---- per-state metadata ----
    const bool odd = (s & 1) != 0;
    const int  j   = (s - 1) >> 1;
    const int  jc  = j     < 0 ? 0 : (j     > S - 1 ? S - 1 : j);
    const int  jp  = j - 1 < 0 ? 0 : (j - 1 > S - 1 ? S - 1 : j - 1);
    const int  cls  = odd ? targets[b * S + jc] : 0;
    const int  cprv = (odd && s >= 3) ? targets[b * S + jp] : 0;
    const bool skipf  = odd && (s >= 3) && (cls != cprv);
    const bool validf = s < 2 * tl + 1;

    // ---- prologue: fill async ring with rows 0..PF-1 ----
#pragma unroll
    for (int r = 0; r < PF; ++r)
        async_row_tile(log_probs + ((size_t)r * B + b) * C, &rows[r * ROWF], off);

    // NEG guard slots for both alpha buffers (written once, never overwritten)
    if (tid < 2) { shA[tid] = NEGV; shA[AP + tid] = NEGV; }

    WAIT_ASYNCCNT(PF - 1);                // own slice of row 0 retired
    __syncthreads();                      // whole row 0 present, guards visible

    // ---- t = 0 init (buffer 0) ----
    float alpha;
    {
        float a = NEGV;
        if (s == 0)      a = rows[0];
        else if (s == 1) a = rows[cls];
        alpha = validf ? a : NEGV;
        shA[s + 2] = alpha;
    }

    // ---- main scan t = 1 .. min(te, T-PF) ----
#pragma unroll 4
    for (int t = 1; t <= tmain; ++t) {
        WAIT_DSCNT(0);                    // alpha store (and init) retired
        BAR_SIGNAL();
        WAIT_ASYNCCNT(PF - 2);            // own slice of row t (overlaps barrier)
        BAR_WAIT();

        // slot (t-1)%PF proven free -> refill with row t-1+PF (<= T-1 here)
        async_row_tile(log_probs + ((size_t)(t - 1 + PF) * B + b) * C,
                       &rows[((t - 1) & (PF - 1)) * ROWF], off);

        const float* rowbuf = &rows[(t & (PF - 1)) * ROWF];
        const float* oldA   = &shA[((t - 1) & 1) * AP];
        float lp  = rowbuf[cls];
        float a2v = oldA[s];              // adjacent loads -> ds_load_2addr
        float a1  = oldA[s + 1];
        float a2  = skipf ? a2v : NEGV;

        float comb = lae3(alpha, a1, a2);
        alpha = validf ? (comb + lp) : NEGV;      // t <= te < il: always update
        shA[(t & 1) * AP + s + 2] = alpha;
    }

    // ---- tail t = T-3 .. te (only when te > T-PF; all rows already issued) ----
    if (te > T - PF) {
#pragma unroll
        for (int t = T - PF + 1; t <= T - 1; ++t) {
            if (t > te) break;            // uniform: barriers stay matched
            WAIT_DSCNT(0);
            BAR_SIGNAL();
            const int rem = (T - 1) - t;  // 2,1,0 (compile-time here)
            if (rem >= 2)      { WAIT_ASYNCCNT(2); }
            else if (rem == 1) { WAIT_ASYNCCNT(1); }
            else               { WAIT_ASYNCCNT(0); }
            BAR_WAIT();

            const float* rowbuf = &rows[(t & (PF - 1)) * ROWF];
            const float* oldA   = &shA[((t - 1) & 1) * AP];
            float lp  = rowbuf[cls];
            float a2v = oldA[s];
            float a1  = oldA[s + 1];
            float a2  = skipf ? a2v : NEGV;

            float comb = lae3(alpha, a1, a2);
            alpha = validf ? (comb + lp) : NEGV;
            shA[(t & 1) * AP + s + 2] = alpha;
        }
    }
    __syncthreads();                      // final alpha buffer visible

    // ---- final log-likelihood (thread 0) ----
    if (tid == 0) {
        const int tlast = te > 0 ? te : 0;               // last written step
        const float* fin = &shA[(tlast & 1) * AP];
        const int end = 2 * tl;
        const int e1  = end > 0 ? end - 1 : 0;
        float a_end = fin[end + 2];
        float a_lab = (tl > 0) ? fin[e1 + 2] : NEGV;
        float ll    = lae(a_end, a_lab);
        float loss  = (ll > NEGV * 0.5f) ? -ll : 0.0f;   // zero_infinity
        int   den   = tl > 0 ? tl : 1;
        partial[b]  = loss / (float)den;
    }
}

__global__ __launch_bounds__(32)
void ctc_reduce_mean(const float* __restrict__ partial, float* __restrict__ out) {
    float v = partial[threadIdx.x];                      // B == 32 == wave32
#pragma unroll
    for (int d = 16; d >= 1; d >>= 1) v += __shfl_xor(v, d, 32);
    if (threadIdx.x == 0) out[0] = v / 32.0f;
}

extern "C" void kernel_launch(void* const* d_in, const int* in_sizes, int n_in,
                              void* d_out, int out_size, void* d_ws, size_t ws_size,
                              hipStream_t stream) {
    const float* log_probs      = (const float*)d_in[0];   // (T,B,C) f32
    const int*   targets        = (const int*)  d_in[1];   // (B,S)   i32
    const int*   input_lengths  = (const int*)  d_in[2];   // (B,)    i32
    const int*   target_lengths = (const int*)  d_in[3];   // (B,)    i32
    float* partial = (float*)d_ws;
    float* out     = (float*)d_out;

    ctc_alpha_kernel<<<B, NT, 0, stream>>>(log_probs, targets, input_lengths,
                                           target_lengths, partial);
    ctc_reduce_mean<<<1, 32, 0, stream>>>(partial, out);
}